// FlowReversal_21638045237423
// MI455X (gfx1250) — compile-verified
//
#include <hip/hip_runtime.h>

// Problem dimensions (fixed by the reference).
#define BB 8
#define CC 3
#define HH 544
#define WW 960

static constexpr int HWp  = HH * WW;            // 522240 pixels per plane
static constexpr int PIX  = BB * HWp;           // 4,177,920 source pixels
static constexpr int N1   = BB * CC * HWp;      // 12,533,760 elems per output tensor
static constexpr int NOUT = 2 * N1;             // imgw + o concatenated

typedef float v2f __attribute__((ext_vector_type(2)));

__device__ __forceinline__ void atomic_add_f32(float* p, float v) {
    // Relaxed, device-scope fp32 add; result is dead so the backend selects the
    // non-returning global_atomic_add_f32 (STOREcnt path, fire-and-forget).
    (void)__hip_atomic_fetch_add(p, v, __ATOMIC_RELAXED, __HIP_MEMORY_SCOPE_AGENT);
}

// ---------------------------------------------------------------------------
// Scatter kernel (first in file so the disasm snippet shows it).
// One thread per source pixel. All 4 input loads issued up front (NT hint so
// the 83 MB read-once stream does not evict the L2-resident atomic working
// set), separable Gaussian weights (8 v_exp_f32 instead of 16), 16-neighbor
// splat with 4 L2-resident atomics per valid neighbor (3 img channels + 1
// weight into the channel-0 plane only; replicated afterwards).
// ---------------------------------------------------------------------------
__global__ void fr_scatter_kernel(const float*  __restrict__ img,
                                  const v2f*    __restrict__ flo,
                                  float* __restrict__ imgw,
                                  float* __restrict__ owt) {
    int p = blockIdx.x * blockDim.x + threadIdx.x;
    if (p >= PIX) return;

    int b  = p / HWp;
    int hw = p - b * HWp;
    int base_img = b * CC * HWp + hw;

    // Issue all four loads before any math so their latencies overlap and the
    // compiler can use partial s_wait_loadcnt values.
    v2f   f  = __builtin_nontemporal_load(&flo[p]);            // (B,H,W,2) -> v2f at p
    float i0 = __builtin_nontemporal_load(&img[base_img]);
    float i1 = __builtin_nontemporal_load(&img[base_img + HWp]);
    float i2 = __builtin_nontemporal_load(&img[base_img + 2 * HWp]);

    float x = (f.x + 1.0f) * 0.5f * (float)WW;
    float y = (f.y + 1.0f) * 0.5f * (float)HH;
    float xf = floorf(x);
    float yf = floorf(y);
    int xi0 = (int)xf;
    int yi0 = (int)yf;
    float fx = x - xf;   // in [0,1)
    float fy = y - yf;

    // Separable Gaussian: w(dx,dy) = exp(-dx^2) * exp(-dy^2).
    float ex[4], ey[4];
#pragma unroll
    for (int j = 0; j < 4; ++j) {
        float dxv = fx - (float)(j - 1);   // x - (floor(x) + {-1,0,1,2})
        float dyv = fy - (float)(j - 1);
        ex[j] = __expf(-dxv * dxv);
        ey[j] = __expf(-dyv * dyv);
    }

    int ob = b * CC * HWp;   // base of this batch's channel-0 plane

    // Fast path: entire 4x4 footprint in-bounds (~98% of pixels; border band
    // is only 3 of 544 rows / 3 of 960 cols per side).
    bool interior = (xi0 >= 1) & (xi0 <= WW - 3) & (yi0 >= 1) & (yi0 <= HH - 3);
    if (interior) {
        int idx00 = ob + (yi0 - 1) * WW + (xi0 - 1);
#pragma unroll
        for (int jy = 0; jy < 4; ++jy) {
            int rowbase = idx00 + jy * WW;
            float wy = ey[jy];
#pragma unroll
            for (int jx = 0; jx < 4; ++jx) {
                float w = ex[jx] * wy;
                int idx = rowbase + jx;
                atomic_add_f32(&imgw[idx],           w * i0);
                atomic_add_f32(&imgw[idx +     HWp], w * i1);
                atomic_add_f32(&imgw[idx + 2 * HWp], w * i2);
                atomic_add_f32(&owt[idx], w);
            }
        }
    } else {
        // Border path: per-neighbor validity (skipping == adding zero weight,
        // matching the reference's clip+zero-weight semantics).
#pragma unroll
        for (int jy = 0; jy < 4; ++jy) {
            int yi = yi0 + jy - 1;
            bool vy = (yi >= 0) && (yi < HH);
            int rowbase = ob + yi * WW;
            float wy = ey[jy];
#pragma unroll
            for (int jx = 0; jx < 4; ++jx) {
                int xi = xi0 + jx - 1;
                if (vy && xi >= 0 && xi < WW) {
                    float w = ex[jx] * wy;
                    int idx = rowbase + xi;
                    atomic_add_f32(&imgw[idx],           w * i0);
                    atomic_add_f32(&imgw[idx +     HWp], w * i1);
                    atomic_add_f32(&imgw[idx + 2 * HWp], w * i2);
                    atomic_add_f32(&owt[idx], w);
                }
            }
        }
    }
}

// ---------------------------------------------------------------------------
// Zero the 100 MB output with 16B stores (write-allocates into L2, which is
// exactly where we want the lines resident for the atomic phase).
// ---------------------------------------------------------------------------
__global__ void fr_zero_kernel(float4* __restrict__ out, int n4) {
    int i = blockIdx.x * blockDim.x + threadIdx.x;
    if (i < n4) out[i] = make_float4(0.f, 0.f, 0.f, 0.f);
}

// ---------------------------------------------------------------------------
// Replicate the weight plane from channel 0 to channels 1,2
// (bandwidth-trivial: 17 MB read + 33 MB write, mostly L2 hits).
// ---------------------------------------------------------------------------
__global__ void fr_replicate_kernel(float* __restrict__ owt) {
    int p = blockIdx.x * blockDim.x + threadIdx.x;
    if (p >= PIX) return;
    int b  = p / HWp;
    int hw = p - b * HWp;
    int idx = b * CC * HWp + hw;
    float v = owt[idx];
    owt[idx +     HWp] = v;
    owt[idx + 2 * HWp] = v;
}

extern "C" void kernel_launch(void* const* d_in, const int* in_sizes, int n_in,
                              void* d_out, int out_size, void* d_ws, size_t ws_size,
                              hipStream_t stream) {
    const float* img = (const float*)d_in[0];   // (B,C,H,W) f32
    const v2f*   flo = (const v2f*)d_in[1];     // (B,H,W,2) f32

    float* out  = (float*)d_out;
    float* imgw = out;        // first output tensor
    float* owt  = out + N1;   // second output tensor

    // 1) zero outputs (must happen every call; harness does not re-poison).
    int n4 = NOUT / 4;  // exactly divisible
    fr_zero_kernel<<<n4 / 256, 256, 0, stream>>>((float4*)out, n4);

    // 2) scatter splats (PIX is an exact multiple of 256 = 8 wave32s).
    fr_scatter_kernel<<<PIX / 256, 256, 0, stream>>>(img, flo, imgw, owt);

    // 3) broadcast weight plane across channels.
    fr_replicate_kernel<<<PIX / 256, 256, 0, stream>>>(owt);
}